// VolumeRenderer_79027398246888
// MI455X (gfx1250) — compile-verified
//
#include <hip/hip_runtime.h>
#include <math.h>

#define AS1 __attribute__((address_space(1)))
#define AS3 __attribute__((address_space(3)))

typedef __attribute__((ext_vector_type(4))) int v4i;

// --- CDNA5 async global->LDS path (gfx1250), guarded so we compile either way ---
#if defined(__HIP_DEVICE_COMPILE__)
  #if __has_builtin(__builtin_amdgcn_global_load_async_to_lds_b128)
    #define HAS_ASYNC_LDS 1
  #else
    #define HAS_ASYNC_LDS 0
  #endif
  #if __has_builtin(__builtin_amdgcn_s_wait_asynccnt)
    #define WAIT_ASYNC(n) __builtin_amdgcn_s_wait_asynccnt(n)
  #else
    #define WAIT_ASYNC(n) asm volatile("s_wait_asynccnt " #n ::: "memory")
  #endif
#else
  #define HAS_ASYNC_LDS 0
  #define WAIT_ASYNC(n)
#endif

constexpr int NS   = 128;        // samples per ray (fixed by reference)
constexpr int WPB  = 8;          // waves per block (256 threads, wave32)
constexpr int RPW  = 8;          // rays per wave (pipelined)
constexpr int RGBF = NS * 3;     // 384 floats of rgb per ray (1536 B)

#if HAS_ASYNC_LDS
__device__ __forceinline__ void prefetch_rgb48(const float* gsrc, float* lbuf) {
  // 48 bytes per lane = 3 x b128; INST_OFFSET is added to BOTH global and LDS addr.
  AS1 v4i* g = (AS1 v4i*)(AS1 const float*)gsrc;
  AS3 v4i* l = (AS3 v4i*)(AS3 float*)lbuf;
  __builtin_amdgcn_global_load_async_to_lds_b128(g, l, 0,  0);
  __builtin_amdgcn_global_load_async_to_lds_b128(g, l, 16, 0);
  __builtin_amdgcn_global_load_async_to_lds_b128(g, l, 32, 0);
}
#endif

__global__ __launch_bounds__(WPB * 32, 1) void volrend_kernel(
    const float* __restrict__ rgb,
    const float* __restrict__ density,
    const float* __restrict__ z_vals,
    const float* __restrict__ rays_d,
    float* __restrict__ o_rgb,
    float* __restrict__ o_depth,
    float* __restrict__ o_w,
    int n_rays)
{
#if HAS_ASYNC_LDS
  __shared__ __align__(16) float smem[WPB * 2 * RGBF];   // 24 KB: 2 buffers / wave
#endif
  const int lane = threadIdx.x & 31;
  const int wib  = threadIdx.x >> 5;
  const long ray0 = ((long)blockIdx.x * WPB + wib) * RPW;

#if HAS_ASYNC_LDS
  float* bufA = &smem[wib * 2 * RGBF];
  float* bufB = bufA + RGBF;
  if (ray0 < n_rays)
    prefetch_rgb48(rgb + (size_t)ray0 * RGBF + lane * 12, bufA + lane * 12);
#endif

  for (int i = 0; i < RPW; ++i) {
    const long ray = ray0 + i;
    if (ray >= n_rays) break;

    // ---- coalesced 16B loads: 4 samples per lane ----
    const float4 zv = ((const float4*)(z_vals  + (size_t)ray * NS))[lane];
    const float4 dn = ((const float4*)(density + (size_t)ray * NS))[lane];
    const float rdx = rays_d[ray * 3 + 0];
    const float rdy = rays_d[ray * 3 + 1];
    const float rdz = rays_d[ray * 3 + 2];
    const float nrm = sqrtf(rdx * rdx + rdy * rdy + rdz * rdz);

    // dists (last interval = 1e10), scaled by |rays_d|
    const float znext = __shfl_down(zv.x, 1, 32);       // z[4L+4] from lane L+1
    const float d0 = (zv.y - zv.x) * nrm;
    const float d1 = (zv.z - zv.y) * nrm;
    const float d2 = (zv.w - zv.z) * nrm;
    const float d3 = (lane == 31) ? (1e10f * nrm) : ((znext - zv.w) * nrm);

    // alpha_i = 1 - exp(-relu(sigma)*dist);  t_i = (1-alpha)+1e-10
    const float e0 = expf(-fmaxf(dn.x, 0.f) * d0);
    const float e1 = expf(-fmaxf(dn.y, 0.f) * d1);
    const float e2 = expf(-fmaxf(dn.z, 0.f) * d2);
    const float e3 = expf(-fmaxf(dn.w, 0.f) * d3);
    const float a0 = 1.f - e0, a1 = 1.f - e1, a2 = 1.f - e2, a3 = 1.f - e3;
    const float t0 = e0 + 1e-10f, t1 = e1 + 1e-10f, t2 = e2 + 1e-10f, t3 = e3 + 1e-10f;

    // in-lane exclusive prefix products + multiplicative wave scan (5 steps)
    const float p1 = t0, p2 = t0 * t1, p3 = p2 * t2;
    float sc = p3 * t3;                                  // lane-total product
    #pragma unroll
    for (int off = 1; off < 32; off <<= 1) {
      const float v = __shfl_up(sc, off, 32);
      if (lane >= off) sc *= v;
    }
    float T = __shfl_up(sc, 1, 32);                      // exclusive across lanes
    if (lane == 0) T = 1.f;

    const float w0 = a0 * T;
    const float w1 = a1 * T * p1;
    const float w2 = a2 * T * p2;
    const float w3 = a3 * T * p3;

    ((float4*)(o_w + (size_t)ray * NS))[lane] = make_float4(w0, w1, w2, w3);

    // ---- rgb weighted sum: current buffer from LDS, prefetch next ray ----
#if HAS_ASYNC_LDS
    float* cur = (i & 1) ? bufB : bufA;
    if (i + 1 < RPW && ray + 1 < n_rays) {
      float* nxt = (i & 1) ? bufA : bufB;
      prefetch_rgb48(rgb + (size_t)(ray + 1) * RGBF + lane * 12, nxt + lane * 12);
      WAIT_ASYNC(3);   // in-order completion => current buffer's 3 loads are done
    } else {
      WAIT_ASYNC(0);
    }
    const float4* rgb4 = (const float4*)cur + lane * 3;
#else
    const float4* rgb4 = (const float4*)(rgb + (size_t)ray * RGBF) + lane * 3;
#endif
    const float4 c0 = rgb4[0], c1 = rgb4[1], c2 = rgb4[2];
    // sample k colors: s0=(c0.x,c0.y,c0.z) s1=(c0.w,c1.x,c1.y) s2=(c1.z,c1.w,c2.x) s3=(c2.y,c2.z,c2.w)
    float r = w0 * c0.x + w1 * c0.w + w2 * c1.z + w3 * c2.y;
    float g = w0 * c0.y + w1 * c1.x + w2 * c1.w + w3 * c2.z;
    float b = w0 * c0.z + w1 * c1.y + w2 * c2.x + w3 * c2.w;
    float dep = w0 * zv.x + w1 * zv.y + w2 * zv.z + w3 * zv.w;

    #pragma unroll
    for (int off = 16; off > 0; off >>= 1) {
      r   += __shfl_down(r,   off, 32);
      g   += __shfl_down(g,   off, 32);
      b   += __shfl_down(b,   off, 32);
      dep += __shfl_down(dep, off, 32);
    }
    if (lane == 0) {
      o_rgb[ray * 3 + 0] = r;
      o_rgb[ray * 3 + 1] = g;
      o_rgb[ray * 3 + 2] = b;
      o_depth[ray] = dep;
    }
  }
}

extern "C" void kernel_launch(void* const* d_in, const int* in_sizes, int n_in,
                              void* d_out, int out_size, void* d_ws, size_t ws_size,
                              hipStream_t stream) {
  const float* rgb     = (const float*)d_in[0];
  const float* density = (const float*)d_in[1];
  const float* z_vals  = (const float*)d_in[2];
  const float* rays_d  = (const float*)d_in[3];
  const int n_rays = in_sizes[3] / 3;          // rays_d is (N,3)

  float* o_rgb   = (float*)d_out;              // (N,3)
  float* o_depth = o_rgb + (size_t)n_rays * 3; // (N,)
  float* o_w     = o_depth + n_rays;           // (N,128)

  const int rays_per_block = WPB * RPW;        // 64
  const int grid = (n_rays + rays_per_block - 1) / rays_per_block;
  volrend_kernel<<<grid, WPB * 32, 0, stream>>>(rgb, density, z_vals, rays_d,
                                                o_rgb, o_depth, o_w, n_rays);
}